// EdgeCondGCNNTF_60722247631701
// MI455X (gfx1250) — compile-verified
//
#include <hip/hip_runtime.h>
#include <hip/hip_bf16.h>
#include <math.h>

// ---------------------------------------------------------------------------
// EdgeCondGCN on MI455X (gfx1250, wave32).
// Factorized algorithm (avoids the 134MB edge_w tensor, halves FLOPs):
//   xn   = GraphNorm(x)                                  (B,T,N,FIN)
//   h    = gelu(adj*ew1 + eb1)                           (B,N,N,E)
//   W2T[f][(e,o)] = ew2[e][f*64+o]                       (FIN, E*FOUT)
//   P[b,t][j][(e,o)] = xn[b,t] @ W2T                     64x64 @ 64x1024
//   agg[b,t][i][o]   = h[b][i][(j,e)] @ P[b,t][(j,e)][o] 64x1024 @ 1024x64
//   y = gelu(agg + xn@nw + nb + S@eb2)
// Both GEMMs on V_WMMA_F32_16X16X4_F32 (fp32-exact matrix pipe).
// ---------------------------------------------------------------------------

typedef __attribute__((ext_vector_type(2))) float v2f;
typedef __attribute__((ext_vector_type(8))) float v8f;

#define Bb   2
#define Tt   32
#define Nn   64
#define FIN  64
#define FOUT 64
#define EDIM 16
#define BT   (Bb*Tt)        // 64
#define KP   (Nn*EDIM)      // 1024 = contraction (j,e) / columns (e,o)
#define LDA  68             // padded LDS row stride: bank = (4m+k)%64, conflict-free

__device__ __forceinline__ float gelu_exact(float v) {
    return 0.5f * v * (1.0f + erff(v * 0.70710678118654752440f));
}

// ---- 1) GraphNorm over node axis + per-feature sums S -----------------------
__global__ __launch_bounds__(64) void k_norm(const float* __restrict__ x,
                                             const float* __restrict__ gw,
                                             const float* __restrict__ gb,
                                             const float* __restrict__ gms,
                                             float* __restrict__ xn,
                                             float* __restrict__ S) {
    const int bt = blockIdx.x;                 // (b,t)
    const int f  = threadIdx.x;                // feature
    const float* xp = x + (size_t)bt * Nn * FIN;
    float s = 0.f, s2 = 0.f;
    #pragma unroll 8
    for (int j = 0; j < Nn; ++j) { float v = xp[j * FIN + f]; s += v; s2 += v * v; }
    const float mean = s * (1.0f / Nn);
    const float var  = s2 * (1.0f / Nn) - mean * mean;   // biased, matches jnp.var
    const float rstd = rsqrtf(var + 1e-5f);
    const float sub  = mean * gms[0];
    const float w = gw[f], bia = gb[f];
    float* xo = xn + (size_t)bt * Nn * FIN;
    float ssum = 0.f;
    #pragma unroll 8
    for (int j = 0; j < Nn; ++j) {
        float v = (xp[j * FIN + f] - sub) * rstd * w + bia;
        xo[j * FIN + f] = v;
        ssum += v;
    }
    S[bt * FIN + f] = ssum;
}

// ---- 2) h[b][i][j][e] = gelu(adj*ew1[e] + eb1[e]) ---------------------------
__global__ __launch_bounds__(256) void k_h(const float* __restrict__ adj,
                                           const float* __restrict__ ew1,
                                           const float* __restrict__ eb1,
                                           float* __restrict__ h) {
    const int idx = blockIdx.x * 256 + threadIdx.x;     // B*N*N*E = 131072
    const int e  = idx & (EDIM - 1);
    const int ij = idx >> 4;                            // b*N*N + i*N + j
    const float v = adj[ij] * ew1[e] + eb1[e];
    h[idx] = gelu_exact(v);
}

// ---- 3) W2T[f][e*64+o] = ew2[e][f*64+o] -------------------------------------
__global__ __launch_bounds__(256) void k_w2t(const float* __restrict__ ew2,
                                             float* __restrict__ w2t) {
    const int idx = blockIdx.x * 256 + threadIdx.x;     // FIN*E*FOUT = 65536
    const int o = idx & (FOUT - 1);
    const int e = (idx >> 6) & (EDIM - 1);
    const int f = idx >> 10;
    w2t[idx] = ew2[e * (FIN * FOUT) + f * FOUT + o];
}

// ---- 4) biasagg[b,t][o] = sum_f S[b,t,f]*eb2[f][o] --------------------------
__global__ __launch_bounds__(64) void k_biasagg(const float* __restrict__ S,
                                                const float* __restrict__ eb2,
                                                float* __restrict__ biasagg) {
    const int bt = blockIdx.x, o = threadIdx.x;
    float acc = 0.f;
    #pragma unroll 8
    for (int f = 0; f < FIN; ++f) acc += S[bt * FIN + f] * eb2[f * FOUT + o];
    biasagg[bt * FOUT + o] = acc;
}

// ---- 5) P[b,t] = xn[b,t] (64x64) @ W2T (64x1024) ----------------------------
// A slab (16KB) staged once in LDS (pad 68 -> ds_load_b64 hits all 64 banks);
// each B fragment feeds 4 WMMAs (4 M-tiles). W2T is L2-resident across blocks.
__global__ __launch_bounds__(256) void k_stage3(const float* __restrict__ xn,
                                                const float* __restrict__ w2t,
                                                float* __restrict__ P) {
    __shared__ float sA[Nn * LDA];
    const int bt   = blockIdx.x;
    const float* X = xn + (size_t)bt * Nn * FIN;          // [j][f]
    float* Pout    = P + (size_t)bt * Nn * KP;            // [j][(e,o)]

    for (int idx = threadIdx.x; idx < Nn * FIN; idx += 256) {
        const int r = idx >> 6, c = idx & 63;
        sA[r * LDA + c] = X[idx];
    }
    __syncthreads();

    const int wave = threadIdx.x >> 5;                    // wave32: 8 waves
    const int lane = threadIdx.x & 31;
    const int m16  = lane & 15;
    const int koff = (lane >> 4) << 1;                    // 0 or 2
    const int rowadd = (lane >> 4) << 3;                  // +8 rows for hi half

    for (int nt = 0; nt < 8; ++nt) {
        const int n = wave * 128 + nt * 16 + m16;         // C/D column
        v8f a0c = {}, a1c = {}, a2c = {}, a3c = {};
        #pragma unroll 4
        for (int kb = 0; kb < FIN; kb += 4) {
            const int k = kb + koff;
            v2f bf; bf.x = w2t[(size_t)k * KP + n];
                    bf.y = w2t[(size_t)(k + 1) * KP + n];
            v2f a0 = *(const v2f*)(&sA[(0  + m16) * LDA + k]);
            v2f a1 = *(const v2f*)(&sA[(16 + m16) * LDA + k]);
            v2f a2 = *(const v2f*)(&sA[(32 + m16) * LDA + k]);
            v2f a3 = *(const v2f*)(&sA[(48 + m16) * LDA + k]);
            a0c = __builtin_amdgcn_wmma_f32_16x16x4_f32(false, a0, false, bf, (short)0, a0c, false, false);
            a1c = __builtin_amdgcn_wmma_f32_16x16x4_f32(false, a1, false, bf, (short)0, a1c, false, false);
            a2c = __builtin_amdgcn_wmma_f32_16x16x4_f32(false, a2, false, bf, (short)0, a2c, false, false);
            a3c = __builtin_amdgcn_wmma_f32_16x16x4_f32(false, a3, false, bf, (short)0, a3c, false, false);
        }
        #pragma unroll
        for (int r = 0; r < 8; ++r) {
            const int row = r + rowadd;
            Pout[(size_t)(0  + row) * KP + n] = a0c[r];
            Pout[(size_t)(16 + row) * KP + n] = a1c[r];
            Pout[(size_t)(32 + row) * KP + n] = a2c[r];
            Pout[(size_t)(48 + row) * KP + n] = a3c[r];
        }
    }
}

// ---- 6) agg = H[b] (64x1024) @ P[b,t] (1024x64), + xn@nw, + biases, gelu ----
// 16 output tiles; 8 waves x 2 N-tiles fused in one K loop: each A fragment
// (contiguous even-k float2 -> global_load_b64) feeds two independent WMMA
// accumulator chains, halving A traffic and interleaving the XDL pipeline.
__global__ __launch_bounds__(256) void k_stage4(const float* __restrict__ xn,
                                                const float* __restrict__ h,
                                                const float* __restrict__ P,
                                                const float* __restrict__ nw,
                                                const float* __restrict__ nb,
                                                const float* __restrict__ biasagg,
                                                float* __restrict__ y) {
    const int bt = blockIdx.x;
    const int b  = bt >> 5;                               // T = 32
    const float* H  = h  + (size_t)b  * Nn * KP;          // [i][(j,e)]
    const float* Pb = P  + (size_t)bt * Nn * KP;          // [(j,e)][o]
    const float* X  = xn + (size_t)bt * Nn * FIN;         // [i][f]
    float* Y        = y  + (size_t)bt * Nn * FOUT;

    const int wave  = threadIdx.x >> 5;
    const int lane  = threadIdx.x & 31;
    const int mtile = wave & 3;
    const int m16   = lane & 15;
    const int koff  = (lane >> 4) << 1;
    const int rowadd = (lane >> 4) << 3;
    const int arow  = mtile * 16 + m16;
    const int n0    = ((wave >> 2) << 5) + m16;           // N-tile pair {0,1}|{2,3}
    const int n1    = n0 + 16;

    v8f acc0 = {}, acc1 = {};
    // main contraction over (j,e): K = 1024
    #pragma unroll 8
    for (int kb = 0; kb < KP; kb += 4) {
        const int k = kb + koff;
        v2f a = *(const v2f*)(H + (size_t)arow * KP + k);
        v2f b0; b0.x = Pb[(size_t)k * FOUT + n0];
                b0.y = Pb[(size_t)(k + 1) * FOUT + n0];
        v2f b1; b1.x = Pb[(size_t)k * FOUT + n1];
                b1.y = Pb[(size_t)(k + 1) * FOUT + n1];
        acc0 = __builtin_amdgcn_wmma_f32_16x16x4_f32(false, a, false, b0, (short)0, acc0, false, false);
        acc1 = __builtin_amdgcn_wmma_f32_16x16x4_f32(false, a, false, b1, (short)0, acc1, false, false);
    }
    // fused node_lin: xn[b,t] @ nw, K = 64
    #pragma unroll 4
    for (int kb = 0; kb < FIN; kb += 4) {
        const int k = kb + koff;
        v2f a = *(const v2f*)(X + (size_t)arow * FIN + k);
        v2f b0; b0.x = nw[k * FOUT + n0];
                b0.y = nw[(k + 1) * FOUT + n0];
        v2f b1; b1.x = nw[k * FOUT + n1];
                b1.y = nw[(k + 1) * FOUT + n1];
        acc0 = __builtin_amdgcn_wmma_f32_16x16x4_f32(false, a, false, b0, (short)0, acc0, false, false);
        acc1 = __builtin_amdgcn_wmma_f32_16x16x4_f32(false, a, false, b1, (short)0, acc1, false, false);
    }
    const float badd0 = nb[n0] + biasagg[bt * FOUT + n0];
    const float badd1 = nb[n1] + biasagg[bt * FOUT + n1];
    #pragma unroll
    for (int r = 0; r < 8; ++r) {
        const int row = mtile * 16 + r + rowadd;
        Y[(size_t)row * FOUT + n0] = gelu_exact(acc0[r] + badd0);
        Y[(size_t)row * FOUT + n1] = gelu_exact(acc1[r] + badd1);
    }
}

// ---------------------------------------------------------------------------
extern "C" void kernel_launch(void* const* d_in, const int* in_sizes, int n_in,
                              void* d_out, int out_size, void* d_ws, size_t ws_size,
                              hipStream_t stream) {
    const float* x    = (const float*)d_in[0];
    const float* adj  = (const float*)d_in[1];
    const float* ew1  = (const float*)d_in[2];
    const float* eb1  = (const float*)d_in[3];
    const float* ew2  = (const float*)d_in[4];
    const float* eb2  = (const float*)d_in[5];
    const float* nw   = (const float*)d_in[6];
    const float* nb   = (const float*)d_in[7];
    const float* gw   = (const float*)d_in[8];
    const float* gb   = (const float*)d_in[9];
    const float* gms  = (const float*)d_in[10];
    float* y = (float*)d_out;

    float* ws      = (float*)d_ws;
    float* xn      = ws;                       // 262144
    float* S       = xn + BT * Nn * FIN;       // 4096
    float* biasagg = S + BT * FIN;             // 4096
    float* h       = biasagg + BT * FOUT;      // 131072
    float* w2t     = h + Bb * Nn * Nn * EDIM;  // 65536
    float* P       = w2t + FIN * KP;           // 4194304 (~16 MB)

    k_norm   <<<BT, 64, 0, stream>>>(x, gw, gb, gms, xn, S);
    k_h      <<<(Bb * Nn * Nn * EDIM) / 256, 256, 0, stream>>>(adj, ew1, eb1, h);
    k_w2t    <<<(FIN * KP) / 256, 256, 0, stream>>>(ew2, w2t);
    k_biasagg<<<BT, 64, 0, stream>>>(S, eb2, biasagg);
    k_stage3 <<<BT, 256, 0, stream>>>(xn, w2t, P);
    k_stage4 <<<BT, 256, 0, stream>>>(xn, h, P, nw, nb, biasagg, y);
}